// GraphConvolutionalNetwork_446676598800
// MI455X (gfx1250) — compile-verified
//
#include <hip/hip_runtime.h>
#include <hip/hip_bf16.h>

typedef float v2f __attribute__((ext_vector_type(2)));
typedef float v8f __attribute__((ext_vector_type(8)));

#define N_NODES 10000
#define E_EDGES 163840
#define IN_F    512
#define H_F     512
#define OUT_F   64

// ---------------------------------------------------------------------------
// fp32 GEMM via V_WMMA_F32_16X16X4_F32.
// One wave (32 threads) computes a 16(M) x 64(N) strip of C = A @ B.
// A: [M,K] row-major, B: [K,N] row-major, C: [M,N] row-major.
// grid.x = M/16, grid.y = N/64. K % 4 == 0, M % 16 == 0, N % 64 == 0.
// ---------------------------------------------------------------------------
__global__ __launch_bounds__(32) void gemm_f32_wmma(
    const float* __restrict__ A, const float* __restrict__ B,
    float* __restrict__ C, int K, int N) {
  const int lane = threadIdx.x & 31;
  const int r    = lane & 15;        // row-in-tile (A,M) / col-in-tile (B,C)
  const int hi   = lane >> 4;        // lane half selects K pair / M half
  const int m0   = blockIdx.x * 16;
  const int n0   = blockIdx.y * 64;

  v8f acc0 = {}, acc1 = {}, acc2 = {}, acc3 = {};
  const float* arow = A + (size_t)(m0 + r) * K;

  for (int k0 = 0; k0 < K; k0 += 4) {
    const int ka = k0 + 2 * hi;                     // this half-lane's K base
    // A 16x4 fragment: V0 = K=ka, V1 = K=ka+1 (contiguous 8B load)
    const v2f a = *reinterpret_cast<const v2f*>(arow + ka);
    // B 4x16 fragments per n-tile: V0 = row ka, V1 = row ka+1, col = n0+16j+r
    const float* b0 = B + (size_t)ka * N + n0 + r;
    const float* b1 = B + (size_t)(ka + 1) * N + n0 + r;
    v2f b;
    b.x = b0[0];  b.y = b1[0];
    acc0 = __builtin_amdgcn_wmma_f32_16x16x4_f32(false, a, false, b, (short)0, acc0, false, false);
    b.x = b0[16]; b.y = b1[16];
    acc1 = __builtin_amdgcn_wmma_f32_16x16x4_f32(false, a, false, b, (short)0, acc1, false, false);
    b.x = b0[32]; b.y = b1[32];
    acc2 = __builtin_amdgcn_wmma_f32_16x16x4_f32(false, a, false, b, (short)0, acc2, false, false);
    b.x = b0[48]; b.y = b1[48];
    acc3 = __builtin_amdgcn_wmma_f32_16x16x4_f32(false, a, false, b, (short)0, acc3, false, false);
  }

  // C/D layout: VGPR v holds row m0 + v + 8*hi, col n0 + 16*j + r
  v8f accs[4] = {acc0, acc1, acc2, acc3};
#pragma unroll
  for (int j = 0; j < 4; ++j) {
    float* cc = C + n0 + 16 * j + r;
#pragma unroll
    for (int v = 0; v < 8; ++v) {
      cc[(size_t)(m0 + v + 8 * hi) * N] = accs[j][v];
    }
  }
}

// ---------------------------------------------------------------------------
// Degree / normalization helpers
// ---------------------------------------------------------------------------
__global__ void init_deg(float* __restrict__ deg, int n) {
  int i = blockIdx.x * blockDim.x + threadIdx.x;
  if (i < n) deg[i] = 1.0f;  // self-loop
}

__global__ void deg_edges(const int* __restrict__ dst, float* __restrict__ deg, int e) {
  int i = blockIdx.x * blockDim.x + threadIdx.x;
  if (i < e) atomicAdd(&deg[dst[i]], 1.0f);
}

__global__ void deg_to_dinv(float* __restrict__ deg, int n) {
  int i = blockIdx.x * blockDim.x + threadIdx.x;
  if (i < n) deg[i] = rsqrtf(deg[i]);
}

// ---------------------------------------------------------------------------
// Edge scatter: out[dst] += dinv[src]*dinv[dst] * H[src], F features.
// One block per edge; src/dst/weight are block-uniform (scalar loads).
// ---------------------------------------------------------------------------
__global__ __launch_bounds__(256) void scatter_edges_512(
    const int* __restrict__ src, const int* __restrict__ dst,
    const float* __restrict__ dinv, const float* __restrict__ H,
    float* __restrict__ out) {
  const int e = blockIdx.x;
  const int s = src[e], d = dst[e];
  const float w = dinv[s] * dinv[d];
  const float* hs = H + (size_t)s * 512;
  float* od = out + (size_t)d * 512;
  int f = threadIdx.x;
  atomicAdd(&od[f],       w * hs[f]);
  atomicAdd(&od[f + 256], w * hs[f + 256]);
}

__global__ __launch_bounds__(64) void scatter_edges_64(
    const int* __restrict__ src, const int* __restrict__ dst,
    const float* __restrict__ dinv, const float* __restrict__ H,
    float* __restrict__ out) {
  const int e = blockIdx.x;
  const int s = src[e], d = dst[e];
  const float w = dinv[s] * dinv[d];
  atomicAdd(&out[(size_t)d * 64 + threadIdx.x],
            w * H[(size_t)s * 64 + threadIdx.x]);
}

// ---------------------------------------------------------------------------
// Layer finalizers: add self-loop term + bias (and ReLU for layer 1)
// ---------------------------------------------------------------------------
__global__ __launch_bounds__(256) void finalize1(
    float* __restrict__ A1, const float* __restrict__ H1,
    const float* __restrict__ dinv, const float* __restrict__ b1, int total) {
  int idx = blockIdx.x * blockDim.x + threadIdx.x;
  if (idx >= total) return;
  int i = idx >> 9;          // /512
  int f = idx & 511;
  float di = dinv[i];
  float v = A1[idx] + di * di * H1[idx] + b1[f];
  A1[idx] = v > 0.0f ? v : 0.0f;
}

__global__ __launch_bounds__(256) void finalize2(
    float* __restrict__ O, const float* __restrict__ H2,
    const float* __restrict__ dinv, const float* __restrict__ b2, int total) {
  int idx = blockIdx.x * blockDim.x + threadIdx.x;
  if (idx >= total) return;
  int i = idx >> 6;          // /64
  int f = idx & 63;
  float di = dinv[i];
  O[idx] = O[idx] + di * di * H2[idx] + b2[f];
}

// ---------------------------------------------------------------------------
// log_softmax over rows of 64; one wave32 per row, 8 rows per 256-thread block
// ---------------------------------------------------------------------------
__global__ __launch_bounds__(256) void logsoftmax64(
    const float* __restrict__ in, float* __restrict__ out, int n) {
  const int lane = threadIdx.x & 31;
  const int row  = blockIdx.x * 8 + (threadIdx.x >> 5);
  if (row >= n) return;
  const float* rp = in + (size_t)row * 64;
  float v0 = rp[lane], v1 = rp[lane + 32];
  float m = fmaxf(v0, v1);
#pragma unroll
  for (int off = 16; off > 0; off >>= 1) m = fmaxf(m, __shfl_xor(m, off, 32));
  float s = expf(v0 - m) + expf(v1 - m);
#pragma unroll
  for (int off = 16; off > 0; off >>= 1) s += __shfl_xor(s, off, 32);
  const float lse = m + logf(s);
  out[(size_t)row * 64 + lane]      = v0 - lse;
  out[(size_t)row * 64 + lane + 32] = v1 - lse;
}

// ---------------------------------------------------------------------------
extern "C" void kernel_launch(void* const* d_in, const int* in_sizes, int n_in,
                              void* d_out, int out_size, void* d_ws, size_t ws_size,
                              hipStream_t stream) {
  const float* x    = (const float*)d_in[0];   // [10000, 512]
  const int*   edge = (const int*)d_in[1];     // [2, 163840]
  const float* W1   = (const float*)d_in[2];   // [512, 512]
  const float* b1   = (const float*)d_in[3];   // [512]
  const float* W2   = (const float*)d_in[4];   // [512, 64]
  const float* b2   = (const float*)d_in[5];   // [64]
  float* out = (float*)d_out;                  // [2 * 10000 * 64]

  const int* src = edge;
  const int* dst = edge + E_EDGES;

  // Workspace layout (floats): dinv | H1 | A1 ; H2 aliases H1 (dead after finalize1)
  float* ws   = (float*)d_ws;
  float* dinv = ws;                               // 10240 (padded)
  float* H1   = ws + 10240;                       // 10000*512
  float* A1   = H1 + (size_t)N_NODES * H_F;       // 10000*512
  float* H2   = H1;                               // 10000*64 (reuse)

  // Zero accumulators (graph-capturable memset nodes)
  hipMemsetAsync(A1, 0, (size_t)N_NODES * H_F * sizeof(float), stream);
  hipMemsetAsync(out, 0, (size_t)N_NODES * OUT_F * sizeof(float), stream);

  // deg -> dinv (in place)
  init_deg<<<(N_NODES + 255) / 256, 256, 0, stream>>>(dinv, N_NODES);
  deg_edges<<<(E_EDGES + 255) / 256, 256, 0, stream>>>(dst, dinv, E_EDGES);
  deg_to_dinv<<<(N_NODES + 255) / 256, 256, 0, stream>>>(dinv, N_NODES);

  // Layer 1: H1 = x @ W1 ; A1 = relu(scatter(H1) + dinv^2*H1 + b1)
  gemm_f32_wmma<<<dim3(N_NODES / 16, H_F / 64), 32, 0, stream>>>(x, W1, H1, IN_F, H_F);
  scatter_edges_512<<<E_EDGES, 256, 0, stream>>>(src, dst, dinv, H1, A1);
  finalize1<<<(N_NODES * H_F + 255) / 256, 256, 0, stream>>>(A1, H1, dinv, b1, N_NODES * H_F);

  // Layer 2: H2 = A1 @ W2 ; out[:640000] = scatter(H2) + dinv^2*H2 + b2
  gemm_f32_wmma<<<dim3(N_NODES / 16, OUT_F / 64), 32, 0, stream>>>(A1, W2, H2, H_F, OUT_F);
  scatter_edges_64<<<E_EDGES, 64, 0, stream>>>(src, dst, dinv, H2, out);
  finalize2<<<(N_NODES * OUT_F + 255) / 256, 256, 0, stream>>>(out, H2, dinv, b2, N_NODES * OUT_F);

  // out[640000:] = log_softmax(out[:640000], axis=1)
  logsoftmax64<<<(N_NODES + 7) / 8, 256, 0, stream>>>(out, out + (size_t)N_NODES * OUT_F, N_NODES);
}